// HOLE_33681133535341
// MI455X (gfx1250) — compile-verified
//
#include <hip/hip_runtime.h>
#include <math.h>

typedef float v2f __attribute__((ext_vector_type(2)));
typedef float v8f __attribute__((ext_vector_type(8)));

#define DIM   512
#define WAVES 8           // waves (batch elements) per block
#define AEXT  544         // 512 + 16 wrap elements (padded to 544)
#define BEXT  1024        // full duplicated copy of b

// One wave32 per batch element.
// score[b] = sigmoid( (1/512) * sum_{j,k} a[j]*b[k]*rel[(j-k) mod 512] )
// computed as a 16x16x512 GEMM of shifted copies:
//   A'[sl,t] = a[(t+sl)%512], B'[t,sh] = b[(t-16*sh)%512]
//   D[sl,sh] = q[16*sh+sl] = cross-correlation value; contract with rel.
// LDS holds circularly-extended copies so the unrolled loop needs no index
// math: every ds_load uses a constant immediate offset.
__global__ __launch_bounds__(32 * WAVES) void hole_score_kernel(
    const float* __restrict__ ew, const float* __restrict__ eb,
    const float* __restrict__ rw, const float* __restrict__ rb,
    const int*   __restrict__ x,  const int*   __restrict__ y,
    const int*   __restrict__ r,
    float* __restrict__ out, int batchN)
{
    __shared__ float sA[WAVES][AEXT];
    __shared__ float sB[WAVES][BEXT];
    __shared__ float sR[WAVES][DIM];

    const int lane  = threadIdx.x & 31;
    const int wave  = threadIdx.x >> 5;
    const int batch = blockIdx.x * WAVES + wave;
    const bool active = batch < batchN;   // wave-uniform

    if (active) {
        const int xi = x[batch];
        const int yi = y[batch];
        const int ri = r[batch];
        const float4* awp = (const float4*)ew + (size_t)xi * (DIM / 4);
        const float4* abp = (const float4*)eb + (size_t)xi * (DIM / 4);
        const float4* bwp = (const float4*)ew + (size_t)yi * (DIM / 4);
        const float4* bbp = (const float4*)eb + (size_t)yi * (DIM / 4);
        const float4* rwp = (const float4*)rw + (size_t)ri * (DIM / 4);
        const float4* rbp = (const float4*)rb + (size_t)ri * (DIM / 4);
        float4* dA = (float4*)sA[wave];
        float4* dB = (float4*)sB[wave];
        float4* dR = (float4*)sR[wave];
        #pragma unroll
        for (int i = lane; i < DIM / 4; i += 32) {
            float4 w0 = awp[i], b0 = abp[i];
            float4 va; va.x = w0.x + b0.x; va.y = w0.y + b0.y;
                       va.z = w0.z + b0.z; va.w = w0.w + b0.w;
            dA[i] = va;
            if (i < 4) dA[DIM / 4 + i] = va;      // wrap: aExt[512..527]
            float4 w1 = bwp[i], b1 = bbp[i];
            float4 vb; vb.x = w1.x + b1.x; vb.y = w1.y + b1.y;
                       vb.z = w1.z + b1.z; vb.w = w1.w + b1.w;
            dB[i] = vb;
            dB[DIM / 4 + i] = vb;                 // full duplicate
            float4 w2 = rwp[i], b2 = rbp[i];
            float4 vr; vr.x = w2.x + b2.x; vr.y = w2.y + b2.y;
                       vr.z = w2.z + b2.z; vr.w = w2.w + b2.w;
            dR[i] = vr;
        }
    }
    __syncthreads();

    if (active) {
        // f32 16x16x4 fragment layout (wave32):
        //   lanes 0-15 : row/col = lane,    K = {0,1}
        //   lanes 16-31: row/col = lane-16, K = {2,3}
        const int sl   = lane & 15;          // M (unit shift) / N (group shift)
        const int koff = (lane >> 4) << 1;   // 0 or 2

        // Per-lane base pointers; all loop offsets are compile-time constants.
        const float* pa = sA[wave] + (koff + sl);          // 0..17
        const float* pb = sB[wave] + (koff - (sl << 4) + DIM); // 272..514

        v8f acc0 = {};
        v8f acc1 = {};
        #pragma unroll
        for (int t0 = 0; t0 < DIM; t0 += 8) {
            v2f af0 = { pa[t0],     pa[t0 + 1] };
            v2f bf0 = { pb[t0],     pb[t0 + 1] };
            acc0 = __builtin_amdgcn_wmma_f32_16x16x4_f32(
                false, af0, false, bf0, (short)0, acc0, false, false);
            v2f af1 = { pa[t0 + 4], pa[t0 + 5] };
            v2f bf1 = { pb[t0 + 4], pb[t0 + 5] };
            acc1 = __builtin_amdgcn_wmma_f32_16x16x4_f32(
                false, af1, false, bf1, (short)0, acc1, false, false);
        }
        v8f acc = acc0 + acc1;

        // D[m=sl, n=sh]: this lane holds column n = sl, rows m = mbase + v.
        // rel index = 16*n + m.
        const float* Rv = sR[wave] + (sl << 4) + ((lane >> 4) << 3);
        float partial = 0.0f;
        #pragma unroll
        for (int v = 0; v < 8; ++v)
            partial += acc[v] * Rv[v];

        #pragma unroll
        for (int off = 16; off > 0; off >>= 1)
            partial += __shfl_xor(partial, off, 32);

        if (lane == 0) {
            const float score = partial * (1.0f / (float)DIM);
            out[batch] = 1.0f / (1.0f + __expf(-score));
        }
    }
}

extern "C" void kernel_launch(void* const* d_in, const int* in_sizes, int n_in,
                              void* d_out, int out_size, void* d_ws, size_t ws_size,
                              hipStream_t stream) {
    const float* ew = (const float*)d_in[0];  // entity_w [E, 512]
    const float* eb = (const float*)d_in[1];  // entity_b [E, 512]
    const float* rw = (const float*)d_in[2];  // rel_w    [R, 512]
    const float* rb = (const float*)d_in[3];  // rel_b    [R, 512]
    const int*   x  = (const int*)d_in[4];    // [B]
    const int*   y  = (const int*)d_in[5];    // [B]
    const int*   r  = (const int*)d_in[6];    // [B]
    float* out = (float*)d_out;               // [B] float32

    const int batchN = in_sizes[4];
    const int blocks = (batchN + WAVES - 1) / WAVES;
    hipLaunchKernelGGL(hole_score_kernel, dim3(blocks), dim3(32 * WAVES), 0, stream,
                       ew, eb, rw, rb, x, y, r, out, batchN);
}